// DilationR2_17592186045151
// MI455X (gfx1250) — compile-verified
//
#include <hip/hip_runtime.h>
#include <math.h>

// Morphological dilation, 7x7 per-channel kernel derived from dil_metric.
// out[n,c,h,w] = max_{i,j} ( xpad[n,c,h+i-3,w+j-3] - kern[c,i,j] ), pad value 0.

#define KS     7
#define HALO   3
#define TILE   64
#define LDSW   (TILE + 2 * HALO)   // 70
#define IMG_H  128
#define IMG_W  128
#define NCH    32

#if __has_builtin(__builtin_amdgcn_global_load_async_to_lds_b32)
#define USE_ASYNC_LDS 1
#else
#define USE_ASYNC_LDS 0
#endif

typedef __attribute__((address_space(1))) int GASI;  // global, int-typed for builtin
typedef __attribute__((address_space(3))) int LASI;  // LDS, int-typed for builtin

// Generic->AS casts via integer round-trip (always compiles; flat LDS
// addresses truncate to the LDS byte offset per the CDNA5 aperture rules,
// flat global addresses are identity-mapped).
__device__ __forceinline__ GASI* as_global(const void* p) {
    return (GASI*)(unsigned long long)p;
}
__device__ __forceinline__ LASI* as_lds(const void* p) {
    return (LASI*)(unsigned)(unsigned long long)p;
}

__global__ __launch_bounds__(256) void dilation7x7_kernel(
    const float* __restrict__ x, const float* __restrict__ dil,
    float* __restrict__ out, float constC, float halfP)
{
    __shared__ float tile[LDSW * LDSW];
    __shared__ float kw[KS * KS];

    const int tid = (int)threadIdx.x;
    const int tx0 = (int)blockIdx.x * TILE;
    const int ty0 = (int)blockIdx.y * TILE;
    const int z   = (int)blockIdx.z;          // n*NCH + c
    const int c   = z & (NCH - 1);
    const float* xz = x + (size_t)z * (IMG_H * IMG_W);

    // ---- per-channel 7x7 min-plus kernel: const * ||D_c y||^p ----
    if (tid < KS * KS) {
        const int i = tid / KS;
        const int j = tid - i * KS;
        const float yi = (float)(i - KS / 2);
        const float yj = (float)(j - KS / 2);
        const float d00 = dil[c * 4 + 0], d01 = dil[c * 4 + 1];
        const float d10 = dil[c * 4 + 2], d11 = dil[c * 4 + 3];
        const float a0 = d00 * yi + d01 * yj;
        const float a1 = d10 * yi + d11 * yj;
        const float ss = a0 * a0 + a1 * a1;
        kw[tid] = constC * powf(ss, halfP);
    }

    // ---- stage 70x70 halo tile into LDS (async copies for in-bounds, ----
    // ---- ds-store zeros for the padded region; disjoint addresses)    ----
    for (int f = tid; f < LDSW * LDSW; f += 256) {
        const int ly = f / LDSW;
        const int lx = f - ly * LDSW;
        const int gy = ty0 + ly - HALO;
        const int gx = tx0 + lx - HALO;
        if ((unsigned)gy < (unsigned)IMG_H && (unsigned)gx < (unsigned)IMG_W) {
#if USE_ASYNC_LDS
            __builtin_amdgcn_global_load_async_to_lds_b32(
                as_global(xz + gy * IMG_W + gx), as_lds(&tile[f]),
                /*offset=*/0, /*cpol=*/0);
#else
            tile[f] = xz[gy * IMG_W + gx];
#endif
        } else {
            tile[f] = 0.0f;
        }
    }

#if USE_ASYNC_LDS
#if __has_builtin(__builtin_amdgcn_s_wait_asynccnt)
    __builtin_amdgcn_s_wait_asynccnt(0);
#else
    asm volatile("s_wait_asynccnt 0" ::: "memory");
#endif
#endif
    __syncthreads();

    // hoist the 49 weights into registers (uniform per block)
    float kk[KS * KS];
#pragma unroll
    for (int t = 0; t < KS * KS; ++t) kk[t] = kw[t];

    const int ox = tid & (TILE - 1);   // output column in tile
    const int ry = tid >> 6;           // 0..3 -> row group
    float* outz = out + (size_t)z * (IMG_H * IMG_W);

    // each thread: 16 contiguous rows, processed in 4-row register blocks
    for (int rb = 0; rb < 16; rb += 4) {
        const int oyBase = ry * 16 + rb;

        // 10x7 input window covering 4 stacked 7x7 windows
        float win[10][KS];
#pragma unroll
        for (int i = 0; i < 10; ++i) {
            const float* row = &tile[(oyBase + i) * LDSW + ox];
#pragma unroll
            for (int j = 0; j < KS; ++j) win[i][j] = row[j];
        }

#pragma unroll
        for (int r = 0; r < 4; ++r) {
            float m = -__builtin_inff();
#pragma unroll
            for (int i = 0; i < KS; ++i) {
#pragma unroll
                for (int j = 0; j < KS; ++j) {
                    m = fmaxf(m, win[r + i][j] - kk[i * KS + j]);
                }
            }
            outz[(size_t)(ty0 + oyBase + r) * IMG_W + (tx0 + ox)] = m;
        }
    }
}

extern "C" void kernel_launch(void* const* d_in, const int* in_sizes, int n_in,
                              void* d_out, int out_size, void* d_ws, size_t ws_size,
                              hipStream_t stream) {
    (void)n_in; (void)out_size; (void)d_ws; (void)ws_size;
    const float* x   = (const float*)d_in[0];   // [N,32,128,128] f32
    const float* dil = (const float*)d_in[1];   // [32,2,2] f32
    float* out = (float*)d_out;

    const int N = in_sizes[0] / (NCH * IMG_H * IMG_W);

    // match the reference's Python-double scalar math, then cast to f32
    const double alpha = 0.65, T = 1.0;
    const double p = 2.0 * alpha / (2.0 * alpha - 1.0);       // 4.3333...
    const float halfP  = (float)(p * 0.5);
    const float constC = (float)(T * (2.0 * alpha - 1.0) / pow(2.0 * alpha * T, p));

    dim3 grid(IMG_W / TILE, IMG_H / TILE, N * NCH);
    dilation7x7_kernel<<<grid, dim3(256, 1, 1), 0, stream>>>(x, dil, out, constC, halfP);
}